// _MetaPathGAT_58497454571832
// MI455X (gfx1250) — compile-verified
//
#include <hip/hip_runtime.h>
#include <hip/hip_bf16.h>

typedef __attribute__((ext_vector_type(2))) float v2f;
typedef __attribute__((ext_vector_type(8))) float v8f;

#define N_NODES 50000
#define N_EDGES 800000
#define NEG_SLOPE 0.2f
#define CDIV(a, b) (((a) + (b) - 1) / (b))

// ---------------- WMMA f32 GEMM: C[M,N] = A[M,K] @ B[K,N] ----------------
// One wave (32 lanes) per 16x16 output tile. Requires M%16==0, N%16==0, K%4==0.
// A 16x4 f32 frag: lane<16 -> (M=lane, K=k..k+1), lane>=16 -> (M=lane-16, K=k+2..k+3).
// C frag: VGPR r -> M = r + 8*(lane/16), N = lane%16.
__global__ __launch_bounds__(32)
void gat_wmma_gemm_f32(const float* __restrict__ A, const float* __restrict__ B,
                       float* __restrict__ C, int M, int N, int K) {
  const int lane = threadIdx.x & 31;
  const int half = lane >> 4;
  const int idx  = lane & 15;
  const int row  = blockIdx.x * 16 + idx;   // A row this lane loads
  const int col  = blockIdx.y * 16 + idx;   // B col this lane loads
  v8f acc = {};
  for (int k = 0; k < K; k += 4) {
    const int ka = k + 2 * half;
    v2f a, b;
    a.x = A[(size_t)row * K + ka];
    a.y = A[(size_t)row * K + ka + 1];
    b.x = B[(size_t)ka * N + col];
    b.y = B[(size_t)(ka + 1) * N + col];
    acc = __builtin_amdgcn_wmma_f32_16x16x4_f32(false, a, false, b, (short)0, acc,
                                                false, false);
  }
#pragma unroll
  for (int r = 0; r < 8; ++r) {
    C[(size_t)(blockIdx.x * 16 + r + 8 * half) * N + col] = acc[r];
  }
}

// ---------------- fill ----------------
__global__ void gat_fill_f32(float* __restrict__ p, float v, int n) {
  int t = blockIdx.x * blockDim.x + threadIdx.x;
  if (t < n) p[t] = v;
}

// ---------------- el/er: per (node, head) attention logits ----------------
__global__ void gat_eler(const float* __restrict__ feat, const float* __restrict__ al,
                         const float* __restrict__ ar, float* __restrict__ el,
                         float* __restrict__ er, int N, int H, int D) {
  int t = blockIdx.x * blockDim.x + threadIdx.x;
  if (t >= N * H) return;
  int h = t % H;
  int n = t / H;
  const float* f = feat + ((size_t)n * H + h) * D;
  const float* L = al + (size_t)h * D;
  const float* R = ar + (size_t)h * D;
  float sl = 0.f, sr = 0.f;
  for (int d = 0; d < D; ++d) {
    float v = f[d];
    sl += v * L[d];
    sr += v * R[d];
  }
  el[t] = sl;
  er[t] = sr;
}

// float atomic-max via monotonic int encoding (valid incl. -inf initial value)
__device__ __forceinline__ void atomicMaxF32(float* addr, float val) {
  int iv = __float_as_int(val);
  if (iv >= 0)
    atomicMax((int*)addr, iv);
  else
    atomicMin((unsigned int*)addr, (unsigned int)iv);
}

// ---------------- edge pass 1: e = leakyrelu(el[src]+er[dst]); segment max ----
__global__ void gat_edge_max(const int* __restrict__ src, const int* __restrict__ dst,
                             const float* __restrict__ el, const float* __restrict__ er,
                             float* __restrict__ ebuf, float* __restrict__ m,
                             int E, int H) {
  int t = blockIdx.x * blockDim.x + threadIdx.x;
  if (t >= E * H) return;
  int h = t % H;
  int e = t / H;
  int s = src[e], d = dst[e];
  float v = el[s * H + h] + er[d * H + h];
  v = v > 0.f ? v : NEG_SLOPE * v;
  ebuf[t] = v;
  atomicMaxF32(&m[d * H + h], v);
}

// ---------------- edge pass 2: ee = exp(e - m[dst]); segment sum -------------
__global__ void gat_edge_exp(const int* __restrict__ dst, float* __restrict__ ebuf,
                             const float* __restrict__ m, float* __restrict__ ssum,
                             int E, int H) {
  int t = blockIdx.x * blockDim.x + threadIdx.x;
  if (t >= E * H) return;
  int h = t % H;
  int e = t / H;
  int d = dst[e];
  float ee = __expf(ebuf[t] - m[d * H + h]);
  ebuf[t] = ee;
  atomicAdd(&ssum[d * H + h], ee);
}

// ------- edge pass 3: rst[dst] += feat[src] * (ee / s[dst]); 4 floats/thread --
__global__ void gat_edge_agg(const int* __restrict__ src, const int* __restrict__ dst,
                             const float* __restrict__ feat, const float* __restrict__ ebuf,
                             const float* __restrict__ ssum, float* __restrict__ rst,
                             int E, int H, int D, int D4) {
  int t = blockIdx.x * blockDim.x + threadIdx.x;
  if (t >= E * H * D4) return;
  int d4 = t % D4;
  int rem = t / D4;
  int h = rem % H;
  int e = rem / H;
  int s = src[e], d = dst[e];
  float alpha = ebuf[e * H + h] / ssum[d * H + h];
  const float4 v = *((const float4*)(feat + ((size_t)s * H + h) * D) + d4);
  float* o = rst + ((size_t)d * H + h) * D + d4 * 4;
  atomicAdd(o + 0, v.x * alpha);
  atomicAdd(o + 1, v.y * alpha);
  atomicAdd(o + 2, v.z * alpha);
  atomicAdd(o + 3, v.w * alpha);
}

// ---------------- finalize layer1: h = elu(rst + bias) ----------------
__global__ void gat_finalize_elu(const float* __restrict__ rst, const float* __restrict__ bias,
                                 float* __restrict__ out, int N, int F) {
  int t = blockIdx.x * blockDim.x + threadIdx.x;
  if (t >= N * F) return;
  float v = rst[t] + bias[t % F];
  out[t] = v > 0.f ? v : (__expf(v) - 1.f);
}

// ---------------- finalize layer2: out += bias (in place) ----------------
__global__ void gat_finalize_bias(float* __restrict__ out, const float* __restrict__ bias,
                                  int N, int F) {
  int t = blockIdx.x * blockDim.x + threadIdx.x;
  if (t >= N * F) return;
  out[t] += bias[t % F];
}

extern "C" void kernel_launch(void* const* d_in, const int* in_sizes, int n_in,
                              void* d_out, int out_size, void* d_ws, size_t ws_size,
                              hipStream_t stream) {
  const float* x   = (const float*)d_in[0];
  const int*   src = (const int*)d_in[1];
  const int*   dst = (const int*)d_in[2];
  const float* W1  = (const float*)d_in[3];
  const float* al1 = (const float*)d_in[4];
  const float* ar1 = (const float*)d_in[5];
  const float* b1  = (const float*)d_in[6];
  const float* W2  = (const float*)d_in[7];
  const float* al2 = (const float*)d_in[8];
  const float* ar2 = (const float*)d_in[9];
  const float* b2  = (const float*)d_in[10];
  float* out = (float*)d_out;

  const int N = N_NODES, E = N_EDGES;
  const int H1 = 4, D1 = 16, F1 = 64;   // layer 1: 4 heads x 16
  const int IN = 128;
  const int H2 = 1, D2 = 128, F2 = 128; // layer 2: 1 head x 128

  // ---- workspace layout (floats) ----
  float* w = (float*)d_ws;
  float* h1 = w;                     // [N,64]  persists across layers (12.8 MB)
  float* S  = w + (size_t)N * F1;    // reused scratch region
  // layer-1 views of S
  float* feat1 = S;                  // [N,64]
  float* rst1  = S + 3200000;        // [N,64]
  float* ebuf1 = S + 6400000;        // [E,4]
  float* el1   = S + 9600000;        // [N,4]
  float* er1   = el1 + 200000;
  float* m1    = er1 + 200000;
  float* s1    = m1 + 200000;
  // layer-2 views of S (layer-1 temps dead by then)
  float* feat2 = S;                  // [N,128]
  float* ebuf2 = S + 6400000;        // [E,1]
  float* el2   = S + 7200000;        // [N,1]
  float* er2   = el2 + 50000;
  float* m2    = er2 + 50000;
  float* s2    = m2 + 50000;

  const int TB = 256;
  const float NEG_INF = -__builtin_huge_valf();

  // ================= layer 1 =================
  gat_wmma_gemm_f32<<<dim3(N / 16, F1 / 16), 32, 0, stream>>>(x, W1, feat1, N, F1, IN);
  gat_eler<<<CDIV(N * H1, TB), TB, 0, stream>>>(feat1, al1, ar1, el1, er1, N, H1, D1);
  gat_fill_f32<<<CDIV(N * H1, TB), TB, 0, stream>>>(m1, NEG_INF, N * H1);
  gat_fill_f32<<<CDIV(N * H1, TB), TB, 0, stream>>>(s1, 0.f, N * H1);
  gat_fill_f32<<<CDIV(N * F1, TB), TB, 0, stream>>>(rst1, 0.f, N * F1);
  gat_edge_max<<<CDIV(E * H1, TB), TB, 0, stream>>>(src, dst, el1, er1, ebuf1, m1, E, H1);
  gat_edge_exp<<<CDIV(E * H1, TB), TB, 0, stream>>>(dst, ebuf1, m1, s1, E, H1);
  gat_edge_agg<<<CDIV(E * H1 * (D1 / 4), TB), TB, 0, stream>>>(src, dst, feat1, ebuf1, s1,
                                                              rst1, E, H1, D1, D1 / 4);
  gat_finalize_elu<<<CDIV(N * F1, TB), TB, 0, stream>>>(rst1, b1, h1, N, F1);

  // ================= layer 2 =================
  gat_wmma_gemm_f32<<<dim3(N / 16, F2 / 16), 32, 0, stream>>>(h1, W2, feat2, N, F2, F1);
  gat_eler<<<CDIV(N * H2, TB), TB, 0, stream>>>(feat2, al2, ar2, el2, er2, N, H2, D2);
  gat_fill_f32<<<CDIV(N * H2, TB), TB, 0, stream>>>(m2, NEG_INF, N * H2);
  gat_fill_f32<<<CDIV(N * H2, TB), TB, 0, stream>>>(s2, 0.f, N * H2);
  gat_fill_f32<<<CDIV(N * F2, TB), TB, 0, stream>>>(out, 0.f, N * F2);
  gat_edge_max<<<CDIV(E * H2, TB), TB, 0, stream>>>(src, dst, el2, er2, ebuf2, m2, E, H2);
  gat_edge_exp<<<CDIV(E * H2, TB), TB, 0, stream>>>(dst, ebuf2, m2, s2, E, H2);
  gat_edge_agg<<<CDIV(E * H2 * (D2 / 4), TB), TB, 0, stream>>>(src, dst, feat2, ebuf2, s2,
                                                              out, E, H2, D2, D2 / 4);
  gat_finalize_bias<<<CDIV(N * F2, TB), TB, 0, stream>>>(out, b2, N, F2);
}